// MHCLayerAITER_20555713478900
// MI455X (gfx1250) — compile-verified
//
#include <hip/hip_runtime.h>
#include <hip/hip_bf16.h>
#include <math.h>

typedef __attribute__((ext_vector_type(2))) float v2f;
typedef __attribute__((ext_vector_type(8))) float v8f;

#define EPS_F     1e-6f
#define C_DIM     2048
#define N_STREAM  4
#define LDS_STRIDE 2064   // 2048 + 16 floats pad: stream2 base lands on bank 32 -> no conflicts
#define BLOCK     256

// ---------------------------------------------------------------------------
// Kernel 1: tiny setup — sigmoid(H_pre), 2*sigmoid(H_post), Sinkhorn(exp(H_res))
// ws layout: [0..3]=H_pre_act  [4..7]=H_post_act  [8..23]=M (row major 4x4)
// ---------------------------------------------------------------------------
__global__ void mhc_prep_kernel(const float* __restrict__ H_pre,
                                const float* __restrict__ H_post,
                                const float* __restrict__ H_res,
                                float* __restrict__ ws) {
  if (threadIdx.x == 0) {
    float P[16];
    for (int i = 0; i < 16; ++i) P[i] = expf(H_res[i]);
    for (int it = 0; it < 3; ++it) {
      for (int r = 0; r < 4; ++r) {
        float s = P[r*4+0] + P[r*4+1] + P[r*4+2] + P[r*4+3] + EPS_F;
        for (int c = 0; c < 4; ++c) P[r*4+c] /= s;
      }
      for (int c = 0; c < 4; ++c) {
        float s = P[0*4+c] + P[1*4+c] + P[2*4+c] + P[3*4+c] + EPS_F;
        for (int r = 0; r < 4; ++r) P[r*4+c] /= s;
      }
    }
    for (int i = 0; i < 4; ++i)  ws[i]     = 1.0f / (1.0f + expf(-H_pre[i]));
    for (int i = 0; i < 4; ++i)  ws[4+i]   = 2.0f / (1.0f + expf(-H_post[i]));
    for (int i = 0; i < 16; ++i) ws[8+i]   = P[i];
  }
}

// ---------------------------------------------------------------------------
// Kernel 2: one workgroup per batch row b. HBM traffic = read x once (async
// DMA to LDS) + write out once (~512 MB total -> ~22 us floor @ 23.3 TB/s).
// Stream mixing uses v_wmma_f32_16x16x4_f32 (K = 4 = number of streams, fp32
// exactly like the reference einsum).
// ---------------------------------------------------------------------------
__global__ __launch_bounds__(BLOCK)
void mhc_main_kernel(const float* __restrict__ x,
                     const float* __restrict__ w,
                     const float* __restrict__ ws,
                     float* __restrict__ out) {
  __shared__ float xs[N_STREAM * LDS_STRIDE];  // padded x[b] tile (33,024 B)
  __shared__ float ys[C_DIM];                  // y_norm (fp32)
  __shared__ float sM[16];
  __shared__ float sHpre[4];
  __shared__ float sHpost[4];
  __shared__ float s_sumsq;

  const int tid = threadIdx.x;
  const int b   = blockIdx.x;
  const float* xb = x + (size_t)b * (N_STREAM * C_DIM);

  // ---- async DMA: 32 KB x[b] tile -> LDS (8 x b128 per thread, ASYNCcnt) ----
  const unsigned xs_base = (unsigned)(size_t)(void*)xs;  // group-relative LDS byte addr
#pragma unroll
  for (int k = 0; k < 8; ++k) {
    unsigned m   = (unsigned)tid + (unsigned)k * BLOCK;  // 16B-chunk index, 2048 total
    unsigned j   = m >> 9;                               // stream (512 chunks / stream)
    unsigned c0  = (m & 511u) << 2;                      // first channel of chunk
    unsigned lds = xs_base + (j * LDS_STRIDE + c0) * 4u;
    unsigned gof = m << 4;                               // byte offset into x[b]
    asm volatile("global_load_async_to_lds_b128 %0, %1, %2"
                 :: "v"(lds), "v"(gof), "s"(xb) : "memory");
  }

  // small params while the DMA is in flight
  if (tid < 16) sM[tid] = ws[8 + tid];
  if (tid < 4)  { sHpre[tid] = ws[tid]; sHpost[tid] = ws[4 + tid]; }
  if (tid == 0) s_sumsq = 0.0f;

  asm volatile("s_wait_asynccnt 0" ::: "memory");
  __syncthreads();

  // ---- phase 1: sigmoid-weighted stream aggregation + bf16 RMS statistics ----
  const float h0 = sHpre[0], h1 = sHpre[1], h2 = sHpre[2], h3 = sHpre[3];
  float aggs[8];
  float local = 0.0f;
#pragma unroll
  for (int k = 0; k < 8; ++k) {
    int c = tid + k * BLOCK;
    float a = xs[0 * LDS_STRIDE + c] * h0
            + xs[1 * LDS_STRIDE + c] * h1
            + xs[2 * LDS_STRIDE + c] * h2
            + xs[3 * LDS_STRIDE + c] * h3;
    float ab = __bfloat162float(__float2bfloat16(a));  // RNE round like reference
    aggs[k] = ab;
    local  += ab * ab;
  }
  atomicAdd(&s_sumsq, local);  // ds_add_f32
  __syncthreads();
  const float rinv = rsqrtf(s_sumsq * (1.0f / (float)C_DIM) + EPS_F);

#pragma unroll
  for (int k = 0; k < 8; ++k) {
    int c = tid + k * BLOCK;
    float wb = __bfloat162float(__float2bfloat16(w[c]));
    ys[c] = aggs[k] * rinv * wb;   // fp32, as in reference
  }
  __syncthreads();

  // ---- phase 2: stream mix via WMMA f32 16x16x4 + fused epilogue ----
  // A (16x4): row 0 = M[0], row 1 = M[1], row 8 = M[2], row 9 = M[3], rest 0.
  //   A layout: lanes 0-15 hold {K0,K1} of row (lane&15), lanes 16-31 hold {K2,K3}.
  // B (4x16): lane = column (channel), lane-half selects the K pair from LDS.
  // D (16x16): VGPR r = row r (lanes 0-15) / row r+8 (lanes 16-31), so with the
  //   A arrangement above:  d[0] = mixed stream 0|2,  d[1] = mixed stream 1|3
  //   for the low|high lane half — constant indices, no dynamic extracts.
  const int lane = tid & 31;
  const int wid  = tid >> 5;
  const int half = lane >> 4;        // 0 -> this lane stores streams 0/1, 1 -> 2/3
  const int row  = lane & 15;        // A-matrix row held by this lane
  const int rlow = row & 7;
  const int mi   = ((row >> 3) * 2 + rlow) & 3;  // which M row this A row carries
  const bool va  = (rlow < 2);                   // rows 0,1,8,9 carry M; others 0
  v2f a;
  a.x = va ? sM[mi * 4 + 2 * half + 0] : 0.0f;   // K pair per lane half
  a.y = va ? sM[mi * 4 + 2 * half + 1] : 0.0f;

  const int   i0  = 2 * half;                    // first stream this half stores
  const float hp0 = sHpost[i0], hp1 = sHpost[i0 + 1];
  const int   col = lane & 15;
  float* o0 = out + ((size_t)b * N_STREAM + (size_t)i0) * C_DIM;
  float* o1 = o0 + C_DIM;

#pragma unroll
  for (int t = 0; t < 16; ++t) {
    int cc = wid * 256 + t * 16 + col;
    v2f bb;
    bb.x = xs[(2 * half)     * LDS_STRIDE + cc];
    bb.y = xs[(2 * half + 1) * LDS_STRIDE + cc];
    v8f acc = {};
    // (neg_a, A, neg_b, B, c_mod, C, reuse_a, reuse_b)
    v8f d = __builtin_amdgcn_wmma_f32_16x16x4_f32(false, a, false, bb,
                                                  (short)0, acc, false, false);
    float y = ys[cc];
    o0[cc] = d[0] + hp0 * y;  // mixed[b, i0,   cc] + H_post_act*y_norm
    o1[cc] = d[1] + hp1 * y;  // mixed[b, i0+1, cc] + ...
  }
}

// ---------------------------------------------------------------------------
extern "C" void kernel_launch(void* const* d_in, const int* in_sizes, int n_in,
                              void* d_out, int out_size, void* d_ws, size_t ws_size,
                              hipStream_t stream) {
  const float* x     = (const float*)d_in[0];  // [B, 4, 2048] fp32
  const float* wt    = (const float*)d_in[1];  // [2048]
  const float* hpre  = (const float*)d_in[2];  // [4]
  const float* hpost = (const float*)d_in[3];  // [4]
  const float* hres  = (const float*)d_in[4];  // [4, 4]
  float* out = (float*)d_out;
  float* ws  = (float*)d_ws;

  mhc_prep_kernel<<<1, 32, 0, stream>>>(hpre, hpost, hres, ws);

  const int B = in_sizes[0] / (N_STREAM * C_DIM);  // 8192
  mhc_main_kernel<<<B, BLOCK, 0, stream>>>(x, wt, ws, out);
}